// ProjectionLayer_4355096838593
// MI455X (gfx1250) — compile-verified
//
#include <hip/hip_runtime.h>

#define GRIDSZ 100
#define G2     10000      // GRIDSZ*GRIDSZ
#define NCH    306        // MEG channels
#define NCHP   320        // padded to 20 * 16
#define CTILES 20
#define NB     8
#define NT     500
#define GTILES 625        // G2 / 16

typedef __attribute__((ext_vector_type(2))) float v2f;
typedef __attribute__((ext_vector_type(8))) float v8f;
typedef __attribute__((ext_vector_type(4))) float v4f;
typedef __attribute__((ext_vector_type(4))) int   v4i;

// ---------------------------------------------------------------------------
// Kernel 1: nearest-channel argmin via V_WMMA_F32_16X16X4_F32.
// score[g,c] = |loc_c|^2 - 2*(gx*lx + gy*ly)   (|g|^2 dropped: constant per g)
//   A (16x4):  row g -> {gx, gy, 1, 0}
//   B (4x16):  col c -> {-2lx, -2ly, |loc_c|^2, 0}
// B is staged in LDS pre-arranged per WMMA lane-half so the inner loop is a
// single uniform ds_load_b64 + wmma + cmp/cndmask (no divergent EXEC dance).
// ---------------------------------------------------------------------------
__global__ __launch_bounds__(256) void nearest_idx_kernel(
    const float* __restrict__ locs,   // (B, NCH, 2)
    int* __restrict__ idx_out)        // (B, G2)
{
    // sB[0*NCHP + c] = {-2lx, -2ly}  (read by lanes 0-15:  B rows K=0,1)
    // sB[1*NCHP + c] = {|loc|^2, 0}  (read by lanes 16-31: B rows K=2,3)
    __shared__ v2f sB[2 * NCHP];
    const int b   = blockIdx.y;
    const int tid = threadIdx.x;

    for (int c = tid; c < NCHP; c += 256) {
        float lx = 0.0f, ly = 0.0f, nr = 3.0e38f;  // padding -> never argmin
        if (c < NCH) {
            lx = locs[((size_t)b * NCH + c) * 2 + 0];
            ly = locs[((size_t)b * NCH + c) * 2 + 1];
            nr = lx * lx + ly * ly;
        }
        v2f lo; lo.x = -2.0f * lx; lo.y = -2.0f * ly;
        v2f hi; hi.x = nr;         hi.y = 0.0f;
        sB[c]        = lo;
        sB[NCHP + c] = hi;
    }
    __syncthreads();

    const int wave = tid >> 5;                  // 8 waves per block
    const int gt   = blockIdx.x * 8 + wave;     // grid tile [0, 625)
    if (gt >= GTILES) return;                   // wave-uniform: EXEC stays full

    const int lane = tid & 31;
    const int half = lane >> 4;                 // 0: K=0,1 / rows 0-7 of D
    const int l    = lane & 15;
    const int g0   = gt * 16;

    // A matrix per the 32-bit 16x4 layout: lanes 0-15 K={0,1}, lanes 16-31 K={2,3}
    v2f a;
    if (half == 0) {
        const int g = g0 + l;
        a.x = (float)(g / GRIDSZ);              // mgrid x = g // 100
        a.y = (float)(g % GRIDSZ);              // mgrid y = g %  100
    } else {
        a.x = 1.0f;                             // K=2 -> picks up |loc|^2
        a.y = 0.0f;                             // K=3 -> padding
    }

    // Lane-dependent base pointer: resolves the half0/half1 split once.
    const v2f* __restrict__ bptr = &sB[half * NCHP + l];

    float minval[8];
    int   minidx[8];
#pragma unroll
    for (int r = 0; r < 8; ++r) { minval[r] = 3.4e38f; minidx[r] = 0; }

    for (int ct = 0; ct < CTILES; ++ct) {
        const v2f bm = bptr[ct * 16];           // one ds_load_b64, uniform CF
        v8f acc = {};
        acc = __builtin_amdgcn_wmma_f32_16x16x4_f32(
            /*neg_a=*/false, a, /*neg_b=*/false, bm,
            /*c_mod=*/(short)0, acc, /*reuse_a=*/false, /*reuse_b=*/false);

        // D layout: lane holds column c = ct*16 + l; VGPR r = row (half*8 + r)
        const int c = ct * 16 + l;
#pragma unroll
        for (int r = 0; r < 8; ++r) {
            const float v = acc[r];
            if (v < minval[r]) { minval[r] = v; minidx[r] = c; }
        }
    }

    // min-reduce across the 16 lanes of each half (tie -> lowest channel index)
#pragma unroll
    for (int r = 0; r < 8; ++r) {
        float v  = minval[r];
        int   ix = minidx[r];
#pragma unroll
        for (int off = 8; off >= 1; off >>= 1) {
            const float ov = __shfl_xor(v,  off, 16);
            const int   oi = __shfl_xor(ix, off, 16);
            if (ov < v || (ov == v && oi < ix)) { v = ov; ix = oi; }
        }
        if (l == 0)
            idx_out[(size_t)b * G2 + g0 + half * 8 + r] = ix;
    }
}

// ---------------------------------------------------------------------------
// Kernel 2: the bandwidth-bound gather. One block per (b,t) row: stage the
// 306-float channel row in LDS, gather via int4 index loads (idx is 320 KB,
// L2-resident), write with 128-bit nontemporal stores (output never re-read).
// ---------------------------------------------------------------------------
__global__ __launch_bounds__(256) void gather_kernel(
    const float* __restrict__ data,   // (B, T, NCH)
    const int* __restrict__ idx,      // (B, G2)
    float* __restrict__ out)          // (B, T, G2)
{
    __shared__ float row[NCH];
    const int bt  = blockIdx.x;       // [0, B*T)
    const int b   = bt / NT;
    const int tid = threadIdx.x;

    for (int c = tid; c < NCH; c += 256)
        row[c] = data[(size_t)bt * NCH + c];
    __syncthreads();

    const v4i* __restrict__ iv = (const v4i*)(idx + (size_t)b * G2);
    v4f* __restrict__ ov       = (v4f*)(out + (size_t)bt * G2);

    for (int q = tid; q < G2 / 4; q += 256) {
        const v4i i4 = iv[q];
        v4f v;
        v.x = row[i4.x];
        v.y = row[i4.y];
        v.z = row[i4.z];
        v.w = row[i4.w];
        __builtin_nontemporal_store(v, &ov[q]);
    }
}

extern "C" void kernel_launch(void* const* d_in, const int* in_sizes, int n_in,
                              void* d_out, int out_size, void* d_ws, size_t ws_size,
                              hipStream_t stream) {
    const float* data = (const float*)d_in[0];   // (8, 500, 306) f32
    const float* locs = (const float*)d_in[1];   // (8, 306, 2)   f32
    float* out = (float*)d_out;                  // (8, 500, 100, 100) f32
    int*   idx = (int*)d_ws;                     // 8*10000 ints = 320 KB scratch

    // Kernel 1: 625 grid tiles / 8 waves per block -> 79 blocks, y = batch
    dim3 g1((GTILES + 7) / 8, NB);
    nearest_idx_kernel<<<g1, 256, 0, stream>>>(locs, idx);

    // Kernel 2: one block per (b, t) row
    gather_kernel<<<NB * NT, 256, 0, stream>>>(data, idx, out);
}